// GNN_Network_Mesh_Enhanced_14181982011840
// MI455X (gfx1250) — compile-verified
//
#include <hip/hip_runtime.h>
#include <hip/hip_bf16.h>
#include <math.h>

// ---------------------------------------------------------------------------
// GNN_Network_Mesh_Enhanced forward for MI455X (gfx1250).
// Heavy per-edge fc1 GEMM runs on V_WMMA_F32_16X16X4_F32 (fp32 WMMA: the op is
// HBM/gather bound at ~5 FLOP/byte, so we keep full fp32 precision and still
// exercise the CDNA5 matrix pipe). Everything else is small VALU work.
// ---------------------------------------------------------------------------

#define NN 50000      // nodes
#define NE 800000     // edges
#define NQ 1500       // queries
#define NS 12000      // subset

typedef float v2f __attribute__((ext_vector_type(2)));
typedef float v8f __attribute__((ext_vector_type(8)));

__device__ __forceinline__ float prelu(float x, float a) { return x >= 0.f ? x : a * x; }

// ---------------- d_in index map (assumes jax sorted-dict-key flattening) ----
// 0 x, 1 mask, 2 norm_val, 3 x_query, 4 A_feature_edges, 5 merged_nodes,
// then params leaves, then A_edges, A_edges_c, subset_indices, batch, batch_query, n_nodes
// embed blocks (mlp_lplp): a1,a2,l1.b,l1.w,l2.b,l2.w            (6 leaves)
// pred: a1,a2,l1.b,l1.w,l2.b,l2.w,l3.b,l3.w                     (8 leaves)
// attn: a1,a2,a3, fc.a,fc.l1b,fc.l1w,fc.l2b,fc.l2w,
//       fv.a,fv.l1b,fv.l1w,fv.l2b,fv.l2w, fe.a,fe.l1b,fe.l1w,fe.l2b,fe.l2w,
//       p1.b,p1.w,p2.b,p2.w, pv                                 (23 leaves)
// sa:   a1,a2,a3,a4, fc1.b,fc1.w, fc2.b,fc2.w, fe.b,fe.w, fg.b,fg.w (12 leaves)
#define P_EMB   6
#define P_EMBE  12
#define P_EMBM  18
#define P_PRED  24
#define P_RO1   32
#define P_RO2   55
#define P_SA    78
#define IN_EDGES   198
#define IN_EDGES_C 199
#define IN_SUBSET  200

// ---------------------------------------------------------------------------
__global__ void k_zero(float* p, int n) {
  for (int i = blockIdx.x * blockDim.x + threadIdx.x; i < n; i += gridDim.x * blockDim.x)
    p[i] = 0.f;
}

// ---------------- node embeddings: h (7->30->10) and m (7->30->10) ----------
__global__ void k_embed_nodes(const float* __restrict__ x, const float* __restrict__ mask,
                              const float* __restrict__ nv,
                              const float* hl1b, const float* hl1w, const float* hl2b, const float* hl2w,
                              const float* ha1, const float* ha2,
                              const float* ml1b, const float* ml1w, const float* ml2b, const float* ml2w,
                              const float* ma1, const float* ma2,
                              float* __restrict__ H, float* __restrict__ M) {
  __shared__ float W[1104];
  int tid = threadIdx.x;
  for (int i = tid; i < 210; i += blockDim.x) { W[i] = hl1w[i]; W[550 + i] = ml1w[i]; }
  for (int i = tid; i < 30;  i += blockDim.x) { W[210 + i] = hl1b[i]; W[760 + i] = ml1b[i]; }
  for (int i = tid; i < 300; i += blockDim.x) { W[240 + i] = hl2w[i]; W[790 + i] = ml2w[i]; }
  for (int i = tid; i < 10;  i += blockDim.x) { W[540 + i] = hl2b[i]; W[1090 + i] = ml2b[i]; }
  if (tid == 0) { W[1100] = ha1[0]; W[1101] = ha2[0]; W[1102] = ma1[0]; W[1103] = ma2[0]; }
  __syncthreads();
  int n = blockIdx.x * blockDim.x + tid;
  if (n >= NN) return;
  float nvv = nv[n];
  float in7[7], t30[30];
  // h
  for (int i = 0; i < 6; ++i) in7[i] = x[n * 6 + i];
  in7[6] = nvv;
  for (int j = 0; j < 30; ++j) {
    float s = W[210 + j];
    for (int i = 0; i < 7; ++i) s += in7[i] * W[i * 30 + j];
    t30[j] = prelu(s, W[1100]);
  }
  for (int j = 0; j < 10; ++j) {
    float s = W[540 + j];
    for (int i = 0; i < 30; ++i) s += t30[i] * W[240 + i * 10 + j];
    H[n * 20 + j] = prelu(s, W[1101]);
  }
  for (int j = 10; j < 20; ++j) H[n * 20 + j] = 0.f;
  // m
  for (int i = 0; i < 6; ++i) in7[i] = mask[n * 6 + i];
  in7[6] = nvv;
  for (int j = 0; j < 30; ++j) {
    float s = W[760 + j];
    for (int i = 0; i < 7; ++i) s += in7[i] * W[550 + i * 30 + j];
    t30[j] = prelu(s, W[1102]);
  }
  for (int j = 0; j < 10; ++j) {
    float s = W[1090 + j];
    for (int i = 0; i < 30; ++i) s += t30[i] * W[790 + i * 10 + j];
    M[n * 10 + j] = prelu(s, W[1103]);
  }
}

// ---------------- edge embedding: 1->30->10 ---------------------------------
__global__ void k_embed_edges(const float* __restrict__ afe,
                              const float* l1b, const float* l1w, const float* l2b, const float* l2w,
                              const float* a1p, const float* a2p, float* __restrict__ EF) {
  __shared__ float W[372];
  int tid = threadIdx.x;
  for (int i = tid; i < 30; i += blockDim.x) { W[i] = l1w[i]; W[30 + i] = l1b[i]; }
  for (int i = tid; i < 300; i += blockDim.x) W[60 + i] = l2w[i];
  for (int i = tid; i < 10; i += blockDim.x) W[360 + i] = l2b[i];
  if (tid == 0) { W[370] = a1p[0]; W[371] = a2p[0]; }
  __syncthreads();
  int e = blockIdx.x * blockDim.x + tid;
  if (e >= NE) return;
  float xin = afe[e], a1 = W[370], a2 = W[371];
  float t30[30];
  for (int j = 0; j < 30; ++j) t30[j] = prelu(W[30 + j] + xin * W[j], a1);
  for (int j = 0; j < 10; ++j) {
    float s = W[360 + j];
    for (int i = 0; i < 30; ++i) s += t30[i] * W[60 + i * 10 + j];
    EF[e * 10 + j] = prelu(s, a2);
  }
}

// ---------------- global mean pool: prelu(fglobal h + b).mean(0) ------------
template <int IN_CH>
__global__ void k_gpool(const float* __restrict__ H, const float* fgw, const float* fgb,
                        const float* a3p, float* gsum) {
  __shared__ float part[3];
  int tid = threadIdx.x;
  if (tid < 3) part[tid] = 0.f;
  __syncthreads();
  int n = blockIdx.x * blockDim.x + tid;
  if (n < NN) {
    float a3 = a3p[0];
    for (int j = 0; j < 3; ++j) {
      float s = fgb[j];
      for (int i = 0; i < IN_CH; ++i) s += H[n * 20 + i] * fgw[i * 3 + j];
      atomicAdd(&part[j], prelu(s, a3));
    }
  }
  __syncthreads();
  if (tid < 3) atomicAdd(&gsum[tid], part[tid]);
}

// ---------------- per-edge message kernel: WMMA fc1 + scatter-mean ----------
// one wave per 16-edge tile; u = [h(IN_CH) | m(10) | gr(3) | eh(15) | ef(10)]
template <int IN_CH, bool HAS_EF>
__global__ __launch_bounds__(256) void k_sa_edge(
    const int* __restrict__ edges, const float* __restrict__ H, const float* __restrict__ M,
    const float* __restrict__ EF, const float* __restrict__ pos,
    const float* fc1w, const float* fc1b, const float* few, const float* feb,
    const float* a1p, const float* a4p, const float* gsum,
    float* __restrict__ msg, float* __restrict__ cnt) {
  constexpr int K  = IN_CH + 38;
  constexpr int KC = (K + 3) / 4;
  constexpr int KP = KC * 4;
  __shared__ float Wl[64 * 16];       // fc1 weights padded [K][16] -> [64][16]
  __shared__ float tile[8][16 * 68];  // per-wave 16x(KP) input tile, stride 68 (bank-safe)
  __shared__ int   dstb[8][16];
  __shared__ float b1s[16];
  __shared__ float Wes[60];
  __shared__ float Beb[15];
  __shared__ float sc[5];             // a1, a4, gr.xyz
  int tid = threadIdx.x;
  for (int i = tid; i < 64 * 16; i += 256) {
    int k = i >> 4, n = i & 15;
    Wl[i] = (k < K && n < 15) ? fc1w[k * 15 + n] : 0.f;
  }
  if (tid < 60) Wes[tid] = few[tid];
  if (tid < 15) Beb[tid] = feb[tid];
  if (tid < 16) b1s[tid] = (tid < 15) ? fc1b[tid] : 0.f;
  if (tid == 0) {
    const float invN = 1.f / (float)NN;
    sc[0] = a1p[0]; sc[1] = a4p[0];
    sc[2] = gsum[0] * invN; sc[3] = gsum[1] * invN; sc[4] = gsum[2] * invN;
  }
  __syncthreads();

  int w = tid >> 5;
  int lane = tid & 31;
  int e = lane & 15, half = lane >> 4;
  int eg = (blockIdx.x * 8 + w) * 16 + e;
  float* row = &tile[w][e * 68];
  int src = edges[eg];
  if (half == 0) {
    int dst = edges[NE + eg];
    float rx = pos[3 * dst + 0] - pos[3 * src + 0];
    float ry = pos[3 * dst + 1] - pos[3 * src + 1];
    float rz = pos[3 * dst + 2] - pos[3 * src + 2];
    float d2 = rx * rx + ry * ry + rz * rz;
#pragma unroll
    for (int i = 0; i < IN_CH; ++i) row[i] = H[src * 20 + i];
    float a4 = sc[1];
#pragma unroll
    for (int j = 0; j < 15; ++j) {
      float s = Beb[j] + rx * Wes[j] + ry * Wes[15 + j] + rz * Wes[30 + j] + d2 * Wes[45 + j];
      row[IN_CH + 13 + j] = prelu(s, a4);
    }
    dstb[w][e] = dst;
    atomicAdd(&cnt[dst], 1.f);
  } else {
#pragma unroll
    for (int i = 0; i < 10; ++i) row[IN_CH + i] = M[src * 10 + i];
    row[IN_CH + 10] = sc[2]; row[IN_CH + 11] = sc[3]; row[IN_CH + 12] = sc[4];
    if constexpr (HAS_EF) {
#pragma unroll
      for (int i = 0; i < 10; ++i) row[K - 10 + i] = EF[eg * 10 + i];
    } else {
#pragma unroll
      for (int i = 0; i < 10; ++i) row[K - 10 + i] = 0.f;
    }
#pragma unroll
    for (int i = K; i < KP; ++i) row[i] = 0.f;
  }
  // cross-lane LDS staging -> fragment reads: make DS writes land first
  asm volatile("s_wait_dscnt 0" ::: "memory");

  v8f acc = {0.f, 0.f, 0.f, 0.f, 0.f, 0.f, 0.f, 0.f};
  int kofs = half * 2;
  int col = e;
#pragma unroll
  for (int kc = 0; kc < KC; ++kc) {
    int kb = kc * 4 + kofs;
    v2f a, b;
    a.x = row[kb];            a.y = row[kb + 1];
    b.x = Wl[kb * 16 + col];  b.y = Wl[(kb + 1) * 16 + col];
    acc = __builtin_amdgcn_wmma_f32_16x16x4_f32(false, a, false, b, (short)0, acc, false, false);
  }
  if (col < 15) {
    float a1 = sc[0];
#pragma unroll
    for (int r = 0; r < 8; ++r) {
      float v = prelu(acc[r] + b1s[col], a1);
      atomicAdd(&msg[dstb[w][half * 8 + r] * 16 + col], v);
    }
  }
}

// ---------------- per-node fc2: [h,m,agg,gr] -> 20 --------------------------
template <int IN_CH>
__global__ void k_sa_node(const float* __restrict__ Hin, const float* __restrict__ M,
                          const float* __restrict__ msg, const float* __restrict__ cnt,
                          const float* fc2w, const float* fc2b, const float* a2p,
                          const float* gsum, float* __restrict__ Hout) {
  constexpr int D = IN_CH + 28;
  __shared__ float W[984];
  int tid = threadIdx.x;
  for (int i = tid; i < D * 20; i += blockDim.x) W[i] = fc2w[i];
  for (int i = tid; i < 20; i += blockDim.x) W[960 + i] = fc2b[i];
  if (tid == 0) {
    const float invN = 1.f / (float)NN;
    W[980] = a2p[0];
    W[981] = gsum[0] * invN; W[982] = gsum[1] * invN; W[983] = gsum[2] * invN;
  }
  __syncthreads();
  int n = blockIdx.x * blockDim.x + tid;
  if (n >= NN) return;
  float v[48];
  for (int i = 0; i < IN_CH; ++i) v[i] = Hin[n * 20 + i];
  for (int i = 0; i < 10; ++i) v[IN_CH + i] = M[n * 10 + i];
  float c = cnt[n]; c = c < 1.f ? 1.f : c; float ic = 1.f / c;
  for (int i = 0; i < 15; ++i) v[IN_CH + 10 + i] = msg[n * 16 + i] * ic;
  v[IN_CH + 25] = W[981]; v[IN_CH + 26] = W[982]; v[IN_CH + 27] = W[983];
  float a2 = W[980];
  for (int j = 0; j < 20; ++j) {
    float s = W[960 + j];
    for (int i = 0; i < D; ++i) s += v[i] * W[i * 20 + j];
    Hout[n * 20 + j] = prelu(s, a2);
  }
}

// ---------------- kNN: one wave per query, register top-k + LDS merge -------
template <int KK>
__global__ void k_knn(const float* __restrict__ xq, const float* __restrict__ pos,
                      const int* __restrict__ idxmap, int nc, int* __restrict__ nidx) {
  __shared__ float sd[32 * KK];
  __shared__ int   si[32 * KK];
  __shared__ int   sp[32];
  int q = blockIdx.x, lane = threadIdx.x;
  float qx = xq[3 * q], qy = xq[3 * q + 1], qz = xq[3 * q + 2];
  float d[KK]; int id[KK];
#pragma unroll
  for (int t = 0; t < KK; ++t) { d[t] = 3.4e38f; id[t] = 0; }
  for (int j = lane; j < nc; j += 32) {
    int c = idxmap ? idxmap[j] : j;
    float dx = qx - pos[3 * c], dy = qy - pos[3 * c + 1], dz = qz - pos[3 * c + 2];
    float dd = dx * dx + dy * dy + dz * dz;
    if (dd < d[KK - 1]) {
      float cd = dd; int ci = c;
#pragma unroll
      for (int t = 0; t < KK; ++t) {
        if (cd < d[t]) { float td = d[t]; int ti = id[t]; d[t] = cd; id[t] = ci; cd = td; ci = ti; }
      }
    }
  }
#pragma unroll
  for (int t = 0; t < KK; ++t) { sd[lane * KK + t] = d[t]; si[lane * KK + t] = id[t]; }
  sp[lane] = 0;
  __syncthreads();
  if (lane == 0) {
    for (int r = 0; r < KK; ++r) {
      float best = 3.5e38f; int bl = 0;
      for (int l = 0; l < 32; ++l) {
        int p = sp[l];
        if (p < KK) { float v = sd[l * KK + p]; if (v < best) { best = v; bl = l; } }
      }
      nidx[q * KK + r] = si[bl * KK + sp[bl]];
      sp[bl]++;
    }
  }
}

// ---------------- attention per-(query,neighbor) features -------------------
template <int KK>
__global__ __launch_bounds__(256) void k_attn_feat(
    const int* __restrict__ nidx, const float* __restrict__ xq, const float* __restrict__ pos,
    const float* __restrict__ H,
    const float* few1, const float* feb1, const float* few2, const float* feb2, const float* fea,
    const float* fcw1, const float* fcb1, const float* fcw2, const float* fcb2, const float* fca,
    const float* fvw1, const float* fvb1, const float* fvw2, const float* fvb2, const float* fva,
    const float* pv, const float* a1p, const float* a3p,
    float* __restrict__ VAL, float* __restrict__ AL) {
  const int O_FEW1 = 0, O_FEB1 = 120, O_FEW2 = 150, O_FEB2 = 1050, O_FCW1 = 1080, O_FCB1 = 2610,
            O_FCW2 = 2640, O_FCB2 = 5640, O_FVW1 = 5740, O_FVB1 = 7270, O_FVW2 = 7300,
            O_FVB2 = 10300, O_PV = 10400, O_SC = 10500;
  __shared__ float SW[10505];
  int tid = threadIdx.x;
  auto cp = [&](int off, const float* src, int n) {
    for (int i = tid; i < n; i += 256) SW[off + i] = src[i];
  };
  cp(O_FEW1, few1, 120); cp(O_FEB1, feb1, 30); cp(O_FEW2, few2, 900); cp(O_FEB2, feb2, 30);
  cp(O_FCW1, fcw1, 1530); cp(O_FCB1, fcb1, 30); cp(O_FCW2, fcw2, 3000); cp(O_FCB2, fcb2, 100);
  cp(O_FVW1, fvw1, 1530); cp(O_FVB1, fvb1, 30); cp(O_FVW2, fvw2, 3000); cp(O_FVB2, fvb2, 100);
  cp(O_PV, pv, 100);
  if (tid == 0) { SW[O_SC] = fea[0]; SW[O_SC + 1] = fca[0]; SW[O_SC + 2] = fva[0];
                  SW[O_SC + 3] = a1p[0]; SW[O_SC + 4] = a3p[0]; }
  __syncthreads();
  int t = blockIdx.x * 256 + tid;
  if (t >= NQ * KK) return;
  int q = t / KK;
  int c = nidx[t];
  float dx = xq[3 * q] - pos[3 * c], dy = xq[3 * q + 1] - pos[3 * c + 1],
        dz = xq[3 * q + 2] - pos[3 * c + 2];
  float dist = dx * dx + dy * dy + dz * dz;
  float fin[4] = {dx, dy, dz, dist};
  float h1[30], f[51];
  float ae = SW[O_SC], a3 = SW[O_SC + 4];
  for (int j = 0; j < 30; ++j) {
    float s = SW[O_FEB1 + j];
    for (int i = 0; i < 4; ++i) s += fin[i] * SW[O_FEW1 + i * 30 + j];
    h1[j] = prelu(s, ae);
  }
  for (int j = 0; j < 30; ++j) {
    float s = SW[O_FEB2 + j];
    for (int i = 0; i < 30; ++i) s += h1[i] * SW[O_FEW2 + i * 30 + j];
    f[20 + j] = prelu(s, a3);
  }
  for (int i = 0; i < 20; ++i) f[i] = H[c * 20 + i];
  f[50] = dist;
  float c1[30];
  float ac = SW[O_SC + 1], a1 = SW[O_SC + 3];
  for (int j = 0; j < 30; ++j) {
    float s = SW[O_FCB1 + j];
    for (int i = 0; i < 51; ++i) s += f[i] * SW[O_FCW1 + i * 30 + j];
    c1[j] = prelu(s, ac);
  }
  const float inv = 0.22360679775f;  // 1/sqrt(20)
  for (int hd = 0; hd < 5; ++hd) {
    float accv = 0.f;
    for (int l = 0; l < 20; ++l) {
      int colc = hd * 20 + l;
      float s = SW[O_FCB2 + colc];
      for (int i = 0; i < 30; ++i) s += c1[i] * SW[O_FCW2 + i * 100 + colc];
      accv += SW[O_PV + colc] * s;
    }
    AL[t * 5 + hd] = prelu(accv * inv, a1);
  }
  float av = SW[O_SC + 2];
  for (int j = 0; j < 30; ++j) {
    float s = SW[O_FVB1 + j];
    for (int i = 0; i < 51; ++i) s += f[i] * SW[O_FVW1 + i * 30 + j];
    c1[j] = prelu(s, av);
  }
  for (int colc = 0; colc < 100; ++colc) {
    float s = SW[O_FVB2 + colc];
    for (int i = 0; i < 30; ++i) s += c1[i] * SW[O_FVW2 + i * 100 + colc];
    VAL[t * 100 + colc] = s;
  }
}

// ---------------- attention reduce: softmax over k, head mean, proj ---------
template <int KK>
__global__ void k_attn_reduce(const float* __restrict__ VAL, const float* __restrict__ AL,
                              const float* p1w, const float* p1b, const float* p2w,
                              const float* p2b, const float* a2p, float* __restrict__ E) {
  int q = blockIdx.x * blockDim.x + threadIdx.x;
  if (q >= NQ) return;
  float m[20];
  for (int l = 0; l < 20; ++l) m[l] = 0.f;
  for (int hd = 0; hd < 5; ++hd) {
    float mx = -3.4e38f;
    for (int n = 0; n < KK; ++n) { float v = AL[(q * KK + n) * 5 + hd]; mx = v > mx ? v : mx; }
    float w[KK], s = 0.f;
    for (int n = 0; n < KK; ++n) { w[n] = __expf(AL[(q * KK + n) * 5 + hd] - mx); s += w[n]; }
    float is = 1.f / s;
    for (int n = 0; n < KK; ++n) {
      float al = w[n] * is;
      for (int l = 0; l < 20; ++l) m[l] += al * VAL[(q * KK + n) * 100 + hd * 20 + l];
    }
  }
  for (int l = 0; l < 20; ++l) m[l] *= 0.2f;  // mean over 5 heads
  float a2 = a2p[0], p1[20];
  for (int j = 0; j < 20; ++j) {
    float s = p1b[j];
    for (int i = 0; i < 20; ++i) s += m[i] * p1w[i * 20 + j];
    p1[j] = prelu(s, a2);
  }
  for (int j = 0; j < 10; ++j) {
    float s = p2b[j];
    for (int i = 0; i < 20; ++i) s += p1[i] * p2w[i * 10 + j];
    E[q * 10 + j] = s;
  }
}

// ---------------- final prediction MLP: 20->20->20->3 -----------------------
__global__ void k_pred(const float* __restrict__ E1, const float* __restrict__ E2,
                       const float* l1w, const float* l1b, const float* a1p,
                       const float* l2w, const float* l2b, const float* a2p,
                       const float* l3w, const float* l3b, float* __restrict__ out) {
  int q = blockIdx.x * blockDim.x + threadIdx.x;
  if (q >= NQ) return;
  float z[20], t1[20], t2[20];
  for (int i = 0; i < 10; ++i) { z[i] = E1[q * 10 + i]; z[10 + i] = E2[q * 10 + i]; }
  float a1 = a1p[0], a2 = a2p[0];
  for (int j = 0; j < 20; ++j) {
    float s = l1b[j];
    for (int i = 0; i < 20; ++i) s += z[i] * l1w[i * 20 + j];
    t1[j] = prelu(s, a1);
  }
  for (int j = 0; j < 20; ++j) {
    float s = l2b[j];
    for (int i = 0; i < 20; ++i) s += t1[i] * l2w[i * 20 + j];
    t2[j] = prelu(s, a2);
  }
  for (int j = 0; j < 3; ++j) {
    float s = l3b[j];
    for (int i = 0; i < 20; ++i) s += t2[i] * l3w[i * 3 + j];
    out[q * 3 + j] = s;
  }
}

// ---------------------------------------------------------------------------
extern "C" void kernel_launch(void* const* d_in, const int* in_sizes, int n_in,
                              void* d_out, int out_size, void* d_ws, size_t ws_size,
                              hipStream_t stream) {
  auto F = [&](int i) { return (const float*)d_in[i]; };
  auto I = [&](int i) { return (const int*)d_in[i]; };
  float* ws = (float*)d_ws;
  float* H0   = ws + 0;         // 50000 x 20
  float* H1   = ws + 1000000;   // 50000 x 20
  float* Mb   = ws + 2000000;   // 50000 x 10
  float* EFb  = ws + 2500000;   // 800000 x 10
  float* MSG  = ws + 10500000;  // 50000 x 16
  float* CNT  = ws + 11300000;  // 50000
  float* GS   = ws + 11350000;  // 4
  float* E1   = ws + 11360000;  // 1500 x 10
  float* E2   = ws + 11375000;  // 1500 x 10
  int* NIDX1  = (int*)(ws + 11390000);  // 1500 x 15
  int* NIDX2  = (int*)(ws + 11412500);  // 1500 x 10
  float* VAL1 = ws + 11427500;  // 1500 x 15 x 100
  float* AL1  = ws + 13677500;  // 1500 x 15 x 5
  float* VAL2 = ws + 13790000;  // 1500 x 10 x 100
  float* AL2  = ws + 15290000;  // 1500 x 10 x 5

  const float* pos = F(5);
  dim3 b256(256);

  k_embed_nodes<<<(NN + 255) / 256, b256, 0, stream>>>(
      F(0), F(1), F(2),
      F(P_EMB + 2), F(P_EMB + 3), F(P_EMB + 4), F(P_EMB + 5), F(P_EMB + 0), F(P_EMB + 1),
      F(P_EMBM + 2), F(P_EMBM + 3), F(P_EMBM + 4), F(P_EMBM + 5), F(P_EMBM + 0), F(P_EMBM + 1),
      H0, Mb);
  k_embed_edges<<<(NE + 255) / 256, b256, 0, stream>>>(
      F(4), F(P_EMBE + 2), F(P_EMBE + 3), F(P_EMBE + 4), F(P_EMBE + 5),
      F(P_EMBE + 0), F(P_EMBE + 1), EFb);

  for (int l = 0; l < 10; ++l) {
    int pb = P_SA + 12 * l;
    const float *a1 = F(pb + 0), *a2 = F(pb + 1), *a3 = F(pb + 2), *a4 = F(pb + 3);
    const float *fc1b = F(pb + 4), *fc1w = F(pb + 5), *fc2b = F(pb + 6), *fc2w = F(pb + 7);
    const float *feb = F(pb + 8), *few = F(pb + 9), *fgb = F(pb + 10), *fgw = F(pb + 11);
    const float* Hin = (l % 2 == 0) ? H0 : H1;
    float* Hout = (l % 2 == 0) ? H1 : H0;
    const int* edges = (l % 2 == 0) ? I(IN_EDGES) : I(IN_EDGES_C);
    k_zero<<<512, b256, 0, stream>>>(MSG, 850004);  // MSG + CNT + GS contiguous
    if (l == 0) {
      k_gpool<10><<<(NN + 255) / 256, b256, 0, stream>>>(Hin, fgw, fgb, a3, GS);
      k_sa_edge<10, true><<<NE / 16 / 8, b256, 0, stream>>>(
          edges, Hin, Mb, EFb, pos, fc1w, fc1b, few, feb, a1, a4, GS, MSG, CNT);
      k_sa_node<10><<<(NN + 255) / 256, b256, 0, stream>>>(
          Hin, Mb, MSG, CNT, fc2w, fc2b, a2, GS, Hout);
    } else if (l % 2 == 0) {
      k_gpool<20><<<(NN + 255) / 256, b256, 0, stream>>>(Hin, fgw, fgb, a3, GS);
      k_sa_edge<20, true><<<NE / 16 / 8, b256, 0, stream>>>(
          edges, Hin, Mb, EFb, pos, fc1w, fc1b, few, feb, a1, a4, GS, MSG, CNT);
      k_sa_node<20><<<(NN + 255) / 256, b256, 0, stream>>>(
          Hin, Mb, MSG, CNT, fc2w, fc2b, a2, GS, Hout);
    } else {
      k_gpool<20><<<(NN + 255) / 256, b256, 0, stream>>>(Hin, fgw, fgb, a3, GS);
      k_sa_edge<20, false><<<NE / 16 / 8, b256, 0, stream>>>(
          edges, Hin, Mb, nullptr, pos, fc1w, fc1b, few, feb, a1, a4, GS, MSG, CNT);
      k_sa_node<20><<<(NN + 255) / 256, b256, 0, stream>>>(
          Hin, Mb, MSG, CNT, fc2w, fc2b, a2, GS, Hout);
    }
  }
  // final h is in H0 (layer 9 writes H0)
  k_knn<15><<<NQ, dim3(32), 0, stream>>>(F(3), pos, nullptr, NN, NIDX1);
  k_knn<10><<<NQ, dim3(32), 0, stream>>>(F(3), pos, I(IN_SUBSET), NS, NIDX2);
  {
    int pb = P_RO1;
    k_attn_feat<15><<<(NQ * 15 + 255) / 256, b256, 0, stream>>>(
        NIDX1, F(3), pos, H0,
        F(pb + 15), F(pb + 14), F(pb + 17), F(pb + 16), F(pb + 13),
        F(pb + 5), F(pb + 4), F(pb + 7), F(pb + 6), F(pb + 3),
        F(pb + 10), F(pb + 9), F(pb + 12), F(pb + 11), F(pb + 8),
        F(pb + 22), F(pb + 0), F(pb + 2), VAL1, AL1);
    k_attn_reduce<15><<<(NQ + 255) / 256, b256, 0, stream>>>(
        VAL1, AL1, F(pb + 19), F(pb + 18), F(pb + 21), F(pb + 20), F(pb + 1), E1);
  }
  {
    int pb = P_RO2;
    k_attn_feat<10><<<(NQ * 10 + 255) / 256, b256, 0, stream>>>(
        NIDX2, F(3), pos, H0,
        F(pb + 15), F(pb + 14), F(pb + 17), F(pb + 16), F(pb + 13),
        F(pb + 5), F(pb + 4), F(pb + 7), F(pb + 6), F(pb + 3),
        F(pb + 10), F(pb + 9), F(pb + 12), F(pb + 11), F(pb + 8),
        F(pb + 22), F(pb + 0), F(pb + 2), VAL2, AL2);
    k_attn_reduce<10><<<(NQ + 255) / 256, b256, 0, stream>>>(
        VAL2, AL2, F(pb + 19), F(pb + 18), F(pb + 21), F(pb + 20), F(pb + 1), E2);
  }
  k_pred<<<(NQ + 255) / 256, b256, 0, stream>>>(
      E1, E2, F(P_PRED + 3), F(P_PRED + 2), F(P_PRED + 0),
      F(P_PRED + 5), F(P_PRED + 4), F(P_PRED + 1), F(P_PRED + 7), F(P_PRED + 6),
      (float*)d_out);
  (void)in_sizes; (void)n_in; (void)out_size; (void)ws_size;
}